// CausalSelfAttention_58669253264050
// MI455X (gfx1250) — compile-verified
//
#include <hip/hip_runtime.h>
#include <hip/hip_bf16.h>

typedef __attribute__((ext_vector_type(2))) float v2f;
typedef __attribute__((ext_vector_type(8))) float v8f;

#define WMMA_F32(a, b, c) \
  __builtin_amdgcn_wmma_f32_16x16x4_f32(false, (a), false, (b), (short)0, (c), false, false)

// ---------------------------------------------------------------------------
// Kernel 1 / 4: C[M,N] = A[M,K] @ W[N,K]^T  via V_WMMA_F32_16X16X4_F32
// block tile 64(M) x 128(N), 8 waves (4 M-tiles x 2 N-groups), wave tile 16x64
// KC=32, double-buffered LDS, float4 staging, register-batched fragments.
// mode 0: plain row-major store to C
// mode 1: QKV scatter (N=3072): q->qraw[B,H,S,hd], k->kout, v->vout
// ---------------------------------------------------------------------------
#define KC 32
#define LDA (KC + 4)   // 36 floats: 16B-aligned rows, conflict-free frag reads

__global__ __launch_bounds__(256) void wmma_gemm_xwt(
    const float* __restrict__ A, const float* __restrict__ W,
    float* __restrict__ C, int M, int N, int K, int mode,
    float* __restrict__ qraw, float* __restrict__ kout, float* __restrict__ vout)
{
    __shared__ float As[2][64][LDA];
    __shared__ float Ws[2][128][LDA];

    const int tid  = threadIdx.x;
    const int wave = tid >> 5;
    const int lane = tid & 31;
    const int lm   = lane & 15;
    const int half = lane >> 4;

    const int m0 = blockIdx.y * 64;
    const int n0 = blockIdx.x * 128;
    const int wm = (wave & 3) * 16;   // wave M offset in block tile
    const int wn = (wave >> 2) * 64;  // wave N offset in block tile

    // staging coords (float4 granularity): A tile 64x32 = 512 f4 (2/thread),
    // W tile 128x32 = 1024 f4 (4/thread)
    const int ar0 = tid >> 3,          ac0 = (tid & 7) * 4;
    const int ar1 = (tid + 256) >> 3,  ac1 = ac0;
    int wr[4], wc[4];
    #pragma unroll
    for (int u = 0; u < 4; ++u) { int f = tid + u * 256; wr[u] = f >> 3; wc[u] = (f & 7) * 4; }

    float4 rA[2], rW[4];
    // prologue: load block 0 into regs, store to buffer 0
    rA[0] = *(const float4*)&A[(long)(m0 + ar0) * K + ac0];
    rA[1] = *(const float4*)&A[(long)(m0 + ar1) * K + ac1];
    #pragma unroll
    for (int u = 0; u < 4; ++u)
        rW[u] = *(const float4*)&W[(long)(n0 + wr[u]) * K + wc[u]];
    *(float4*)&As[0][ar0][ac0] = rA[0];
    *(float4*)&As[0][ar1][ac1] = rA[1];
    #pragma unroll
    for (int u = 0; u < 4; ++u) *(float4*)&Ws[0][wr[u]][wc[u]] = rW[u];
    __syncthreads();

    v8f acc[4] = {};
    const int nb = K / KC;

    for (int blk = 0; blk < nb; ++blk) {
        const int cur = blk & 1;
        const bool has_next = (blk + 1 < nb);
        // issue next block's global loads (latency hidden under WMMAs)
        if (has_next) {
            const int kc = (blk + 1) * KC;
            rA[0] = *(const float4*)&A[(long)(m0 + ar0) * K + kc + ac0];
            rA[1] = *(const float4*)&A[(long)(m0 + ar1) * K + kc + ac1];
            #pragma unroll
            for (int u = 0; u < 4; ++u)
                rW[u] = *(const float4*)&W[(long)(n0 + wr[u]) * K + kc + wc[u]];
        }
        // compute on current buffer: 8 chunks of K=4, batched 4 at a time
        #pragma unroll
        for (int t4 = 0; t4 < KC / 4; t4 += 4) {
            v2f af[4], bf[4][4];
            #pragma unroll
            for (int u = 0; u < 4; ++u) {
                const int kb = 4 * (t4 + u) + 2 * half;
                af[u] = *(const v2f*)&As[cur][wm + lm][kb];
            }
            #pragma unroll
            for (int nt = 0; nt < 4; ++nt)
                #pragma unroll
                for (int u = 0; u < 4; ++u) {
                    const int kb = 4 * (t4 + u) + 2 * half;
                    bf[nt][u] = *(const v2f*)&Ws[cur][wn + nt * 16 + lm][kb];
                }
            #pragma unroll
            for (int u = 0; u < 4; ++u)
                #pragma unroll
                for (int nt = 0; nt < 4; ++nt)
                    acc[nt] = WMMA_F32(af[u], bf[nt][u], acc[nt]);
        }
        // store next block into the other buffer
        if (has_next) {
            const int nxt = cur ^ 1;
            *(float4*)&As[nxt][ar0][ac0] = rA[0];
            *(float4*)&As[nxt][ar1][ac1] = rA[1];
            #pragma unroll
            for (int u = 0; u < 4; ++u) *(float4*)&Ws[nxt][wr[u]][wc[u]] = rW[u];
        }
        __syncthreads();
    }

    // epilogue: C layout row M=v+8*half, col N=lane%16
    #pragma unroll
    for (int nt = 0; nt < 4; ++nt) {
        #pragma unroll
        for (int vv = 0; vv < 8; ++vv) {
            const int gm = m0 + wm + vv + 8 * half;   // row = b*S + s
            const int gn = n0 + wn + nt * 16 + lm;    // col
            const float val = acc[nt][vv];
            if (mode == 0) {
                C[(long)gm * N + gn] = val;
            } else {
                const int b = gm >> 11, s = gm & 2047;
                const int which = gn >> 10;           // 0=q, 1=k, 2=v
                const int hn = gn & 1023;
                const int h = hn >> 6, d = hn & 63;
                const long idx = ((((long)b * 16 + h) * 2048 + s) * 64 + d);
                if (which == 0)      qraw[idx] = val;
                else if (which == 1) kout[idx] = val;
                else                 vout[idx] = val;
            }
        }
    }
}

// ---------------------------------------------------------------------------
// Kernel 2: RoPE in place on q (ws) and k (d_out), layout [BH, S, 64]
// pair (d, d+32), angle = s * 10000^(-2d/64)
// ---------------------------------------------------------------------------
__global__ __launch_bounds__(256) void rope_kernel(float* __restrict__ q,
                                                   float* __restrict__ k)
{
    const int idx = blockIdx.x * blockDim.x + threadIdx.x;  // BH*S*32 total
    const int d  = idx & 31;
    const int s  = (idx >> 5) & 2047;
    const int bh = idx >> 16;
    if (bh >= 32) return;

    const float inv_freq = __powf(10000.0f, -(float)(2 * d) * (1.0f / 64.0f));
    const float ang = (float)s * inv_freq;
    const float c = __cosf(ang);
    const float sn = __sinf(ang);

    const long base = ((long)bh * 2048 + s) * 64;
    float q1 = q[base + d], q2 = q[base + d + 32];
    q[base + d]      = q1 * c - q2 * sn;
    q[base + d + 32] = q2 * c + q1 * sn;
    float k1 = k[base + d], k2 = k[base + d + 32];
    k[base + d]      = k1 * c - k2 * sn;
    k[base + d + 32] = k2 * c + k1 * sn;
}

// ---------------------------------------------------------------------------
// Kernel 3: causal flash attention, one (b,h,64-query-block) per workgroup.
// 128 threads = 4 waves, each wave owns 16 query rows.
// q pre-scaled; K/V blocks in LDS; P via LDS transpose; online softmax.
// Fragment loads batched 4 k-steps at a time to amortize DS waits.
// o written as [B, S, H*hd] so the out-projection GEMM consumes it directly.
// ---------------------------------------------------------------------------
#define LDK 68   // 64 + 4: 16B-aligned rows, conflict-free

__global__ __launch_bounds__(128) void attn_kernel(
    const float* __restrict__ q, const float* __restrict__ k,
    const float* __restrict__ v, float* __restrict__ o)
{
    const int S = 2048, HD = 64;
    __shared__ float Ks[64][LDK];
    __shared__ float Vs[64][LDK];
    __shared__ float Ps[4][16][LDK];

    const int tid  = threadIdx.x;
    const int wave = tid >> 5;
    const int lane = tid & 31;
    const int lm   = lane & 15;
    const int half = lane >> 4;

    const int qb = blockIdx.x & 31;   // query block within head
    const int bh = blockIdx.x >> 5;   // 0..31
    const int b  = bh >> 4, h = bh & 15;
    const float scale = 0.125f;       // 1/sqrt(64)

    // Q a-frags for this wave's 16 rows, pre-scaled by 1/sqrt(hd)
    v2f qa[16];
    const long qrow = ((long)bh * S + qb * 64 + wave * 16 + lm) * HD;
    #pragma unroll
    for (int t = 0; t < 16; ++t) {
        const int kb = 4 * t + 2 * half;
        v2f qv = *(const v2f*)&q[qrow + kb];
        qa[t].x = qv.x * scale;
        qa[t].y = qv.y * scale;
    }

    float rmax[8], rsum[8];
    v8f o_acc[4] = {};
    #pragma unroll
    for (int i = 0; i < 8; ++i) { rmax[i] = -1e30f; rsum[i] = 0.0f; }

    const long kvbase = (long)bh * S * HD;

    for (int j = 0; j <= qb; ++j) {
        // stage K,V block j (64x64 each): 1024 f4 per matrix, 8/thread
        const float4* kg = (const float4*)(k + kvbase + (long)j * 64 * HD);
        const float4* vg = (const float4*)(v + kvbase + (long)j * 64 * HD);
        #pragma unroll
        for (int u = 0; u < 8; ++u) {
            const int f = tid + u * 128;
            const int r = f >> 4, c4 = (f & 15) * 4;
            *(float4*)&Ks[r][c4] = kg[f];
            *(float4*)&Vs[r][c4] = vg[f];
        }
        __syncthreads();

        // S = Q @ K^T : B[kd][key] = Ks[key][kd]; batched 4 k-steps
        v8f sc[4] = {};
        #pragma unroll
        for (int t4 = 0; t4 < 16; t4 += 4) {
            v2f bf[4][4];
            #pragma unroll
            for (int nt = 0; nt < 4; ++nt)
                #pragma unroll
                for (int u = 0; u < 4; ++u) {
                    const int kb = 4 * (t4 + u) + 2 * half;
                    bf[nt][u] = *(const v2f*)&Ks[nt * 16 + lm][kb];
                }
            #pragma unroll
            for (int u = 0; u < 4; ++u)
                #pragma unroll
                for (int nt = 0; nt < 4; ++nt)
                    sc[nt] = WMMA_F32(qa[t4 + u], bf[nt][u], sc[nt]);
        }

        // causal mask only on the diagonal block
        if (j == qb) {
            #pragma unroll
            for (int nt = 0; nt < 4; ++nt)
                #pragma unroll
                for (int vv = 0; vv < 8; ++vv) {
                    const int qi = wave * 16 + vv + 8 * half;
                    const int kj = nt * 16 + lm;
                    if (kj > qi) sc[nt][vv] = -1e30f;
                }
        }

        // row max across 4 subtiles + 16 lanes of each half
        float bm[8];
        #pragma unroll
        for (int vv = 0; vv < 8; ++vv)
            bm[vv] = fmaxf(fmaxf(sc[0][vv], sc[1][vv]), fmaxf(sc[2][vv], sc[3][vv]));
        #pragma unroll
        for (int off = 8; off >= 1; off >>= 1)
            #pragma unroll
            for (int vv = 0; vv < 8; ++vv)
                bm[vv] = fmaxf(bm[vv], __shfl_xor(bm[vv], off, 32));

        float alpha[8], bs[8];
        #pragma unroll
        for (int vv = 0; vv < 8; ++vv) {
            const float nm = fmaxf(rmax[vv], bm[vv]);
            alpha[vv] = __expf(rmax[vv] - nm);
            rmax[vv] = nm;
            bs[vv] = 0.0f;
        }
        #pragma unroll
        for (int nt = 0; nt < 4; ++nt)
            #pragma unroll
            for (int vv = 0; vv < 8; ++vv) {
                const float p = __expf(sc[nt][vv] - rmax[vv]);
                sc[nt][vv] = p;
                bs[vv] += p;
            }
        #pragma unroll
        for (int off = 8; off >= 1; off >>= 1)
            #pragma unroll
            for (int vv = 0; vv < 8; ++vv)
                bs[vv] += __shfl_xor(bs[vv], off, 32);
        #pragma unroll
        for (int vv = 0; vv < 8; ++vv)
            rsum[vv] = rsum[vv] * alpha[vv] + bs[vv];
        #pragma unroll
        for (int nt = 0; nt < 4; ++nt)
            #pragma unroll
            for (int vv = 0; vv < 8; ++vv)
                o_acc[nt][vv] *= alpha[vv];

        // C-layout -> A-layout via LDS
        #pragma unroll
        for (int nt = 0; nt < 4; ++nt)
            #pragma unroll
            for (int vv = 0; vv < 8; ++vv)
                Ps[wave][vv + 8 * half][nt * 16 + lm] = sc[nt][vv];
        __syncthreads();

        // O += P @ V : B[kk][d] = Vs[kk][d]; batched 4 k-steps
        #pragma unroll
        for (int t4 = 0; t4 < 16; t4 += 4) {
            v2f pa[4], vb[4][4];
            #pragma unroll
            for (int u = 0; u < 4; ++u) {
                const int kb = 4 * (t4 + u) + 2 * half;
                pa[u] = *(const v2f*)&Ps[wave][lm][kb];
            }
            #pragma unroll
            for (int nt = 0; nt < 4; ++nt)
                #pragma unroll
                for (int u = 0; u < 4; ++u) {
                    const int kb = 4 * (t4 + u) + 2 * half;
                    v2f t;
                    t.x = Vs[kb][nt * 16 + lm];
                    t.y = Vs[kb + 1][nt * 16 + lm];
                    vb[nt][u] = t;
                }
            #pragma unroll
            for (int u = 0; u < 4; ++u)
                #pragma unroll
                for (int nt = 0; nt < 4; ++nt)
                    o_acc[nt] = WMMA_F32(pa[u], vb[nt][u], o_acc[nt]);
        }
        __syncthreads();
    }

    // normalize + store as [B, S, H*64]
    #pragma unroll
    for (int vv = 0; vv < 8; ++vv) {
        const float inv = 1.0f / rsum[vv];
        const int s = qb * 64 + wave * 16 + vv + 8 * half;
        const long base = ((long)b * S + s) * 1024 + h * 64;
        #pragma unroll
        for (int nt = 0; nt < 4; ++nt)
            o[base + nt * 16 + lm] = o_acc[nt][vv] * inv;
    }
}

// ---------------------------------------------------------------------------
extern "C" void kernel_launch(void* const* d_in, const int* in_sizes, int n_in,
                              void* d_out, int out_size, void* d_ws, size_t ws_size,
                              hipStream_t stream) {
    (void)in_sizes; (void)n_in; (void)out_size; (void)ws_size;
    const float* x    = (const float*)d_in[0];   // [2,2048,1024]
    const float* Wqkv = (const float*)d_in[1];   // [3072,1024]
    const float* Wout = (const float*)d_in[2];   // [1024,1024]

    float* out  = (float*)d_out;                 // [2,2048,1024]
    float* kout = out + 4194304;                 // [2,16,2048,64] (RoPE'd)
    float* vout = out + 8388608;                 // [2,16,2048,64]

    float* qraw = (float*)d_ws;                  // 16 MB: q [B,H,S,hd]
    float* attn = qraw + 4194304;                // 16 MB: attn out [B,S,D]

    // 1) qkv = x @ Wqkv^T, scatter q/k/v to head-major
    dim3 g1(3072 / 128, 4096 / 64);
    wmma_gemm_xwt<<<g1, 256, 0, stream>>>(x, Wqkv, nullptr, 4096, 3072, 1024, 1,
                                          qraw, kout, vout);
    // 2) RoPE in place on q (ws) and k (d_out)
    rope_kernel<<<8192, 256, 0, stream>>>(qraw, kout);
    // 3) causal flash attention
    attn_kernel<<<1024, 128, 0, stream>>>(qraw, kout, vout, attn);
    // 4) out = attn @ Wout^T
    dim3 g2(1024 / 128, 4096 / 64);
    wmma_gemm_xwt<<<g2, 256, 0, stream>>>(attn, Wout, out, 4096, 1024, 1024, 0,
                                          nullptr, nullptr, nullptr);
}